// FEconvNet_periodicU_H8types_14121852470126
// MI455X (gfx1250) — compile-verified
//
#include <hip/hip_runtime.h>
#include <hip/hip_bf16.h>

typedef __attribute__((ext_vector_type(2))) float v2f;
typedef __attribute__((ext_vector_type(4))) float v4f;
typedef __attribute__((ext_vector_type(8))) float v8f;

#define GN   128              // global grid
#define BN   8                // node tile per block (8^3 = 512 nodes)
#define ET   9                // element halo tile (9^3 = 729 elements)
#define UT   10               // U halo tile (10^3 = 1000 nodes)
#define NELEM (ET*ET*ET)      // 729
#define NROWTILE (ET*ET)      // 81 WMMA tiles: one (ey,ez) row of 9 elements each
#define THREADS 256           // 8 waves (wave32)

// corner k -> offset in U tile (CORNERS order: k = dx + 2*dy + 4*dz)
__device__ __forceinline__ int coff(int k) {
    return (k & 1) + UT * ((k >> 1) & 1) + UT * UT * ((k >> 2) & 1);
}

__global__ __launch_bounds__(THREADS)
void fe_h8_periodic_kernel(const float* __restrict__ U,
                           const int*   __restrict__ types,
                           const float* __restrict__ filters,
                           float*       __restrict__ V) {
    __shared__ float sU[UT * UT * UT];      // 10^3 halo of nodal values
    __shared__ float sM[NELEM];             // element type mask as float
    __shared__ float sVeA[NELEM * 8];       // K0-part of element results
    __shared__ float sVeB[NELEM * 8];       // (K1-K0)-part of element results

    const int tid  = threadIdx.x;
    const int lane = tid & 31;
    const int ln16 = lane & 15;     // A-row / B-col / C-col
    const int half = lane >> 4;     // selects K pair within WMMA operands

    const int ox = blockIdx.x * BN; // node-tile origin (global)
    const int oy = blockIdx.y * BN;
    const int oz = blockIdx.z * BN;

    // ---- Build constant A-matrix operands (16x8 = [K0 ; K1-K0]) -----------
    // 32-bit A 16x4 layout: lane holds M = lane%16; VGPR v holds K = v + 2*(lane/16).
    float a0v[2], a1v[2];
#pragma unroll
    for (int v = 0; v < 2; ++v) {
        const int k0 = v + 2 * half;       // chunk 0: K = 0..3
        const int k1 = 4 + v + 2 * half;   // chunk 1: K = 4..7
        if (ln16 < 8) {                    // rows 0..7  -> K0
            a0v[v] = filters[ln16 * 8 + k0];
            a1v[v] = filters[ln16 * 8 + k1];
        } else {                           // rows 8..15 -> K1 - K0
            const int r = ln16 - 8;
            a0v[v] = filters[64 + r * 8 + k0] - filters[r * 8 + k0];
            a1v[v] = filters[64 + r * 8 + k1] - filters[r * 8 + k1];
        }
    }
    const v2f A0 = {a0v[0], a0v[1]};
    const v2f A1 = {a1v[0], a1v[1]};

    // ---- Phase 1: stage U halo (10^3) and element types (9^3) into LDS ----
    for (int i = tid; i < UT * UT * UT; i += THREADS) {
        const int x = i % UT, y = (i / UT) % UT, z = i / (UT * UT);
        const int gx = (ox - 1 + x) & (GN - 1);
        const int gy = (oy - 1 + y) & (GN - 1);
        const int gz = (oz - 1 + z) & (GN - 1);
        sU[i] = U[(gz * GN + gy) * GN + gx];
    }
    for (int i = tid; i < NELEM; i += THREADS) {
        const int x = i % ET, y = (i / ET) % ET, z = i / (ET * ET);
        const int gx = (ox - 1 + x) & (GN - 1);
        const int gy = (oy - 1 + y) & (GN - 1);
        const int gz = (oz - 1 + z) & (GN - 1);
        sM[i] = (float)types[(gz * GN + gy) * GN + gx];
    }
    __syncthreads();

    // ---- Phase 2: per-element 16x8 matvec on the matrix unit --------------
    // One tile = one (ey,ez) row of 9 elements (columns ex = ln16, lanes 9..15
    // clamped/predicated). All per-lane address components are loop-invariant;
    // only a wave-uniform row base changes per tile.
    const int wave   = tid >> 5;
    const int exc    = (ln16 < ET) ? ln16 : (ET - 1);  // clamp: keep EXEC full
    const bool evld  = (ln16 < ET);
    const int kb     = 2 * half;  // B 4x16 mirrors A: VGPR v holds K = v + 2*(lane/16)
    const int lo0 = exc + coff(kb + 0);   // per-lane constant gather offsets
    const int lo1 = exc + coff(kb + 1);
    const int lo2 = exc + coff(kb + 4);
    const int lo3 = exc + coff(kb + 5);
    float* const veBase = (half ? sVeB : sVeA);

    for (int t = wave; t < NROWTILE; t += (THREADS / 32)) {
        const int ts   = __builtin_amdgcn_readfirstlane(t);  // wave-uniform -> SALU
        const int ey   = ts % ET;
        const int ez   = ts / ET;
        const int rbase = (ez * UT + ey) * UT;               // scalar row base in sU
        const int ebase = ts * ET;                           // scalar row base in sVe

        v2f B0, B1;
        B0[0] = sU[rbase + lo0];
        B0[1] = sU[rbase + lo1];
        B1[0] = sU[rbase + lo2];
        B1[1] = sU[rbase + lo3];

        v8f c = {};
        c = __builtin_amdgcn_wmma_f32_16x16x4_f32(false, A0, false, B0,
                                                  (short)0, c, false, false);
        c = __builtin_amdgcn_wmma_f32_16x16x4_f32(false, A1, false, B1,
                                                  (short)0, c, false, false);

        if (evld) {
            // C layout: lane holds element column ln16, rows v + 8*half.
            float* dst = veBase + (ebase + ln16) * 8;
            v4f clo = {c[0], c[1], c[2], c[3]};
            v4f chi = {c[4], c[5], c[6], c[7]};
            *(v4f*)(dst)     = clo;   // ds_store_b128
            *(v4f*)(dst + 4) = chi;
        }
    }
    __syncthreads();

    // ---- Phase 3: gather 8 element contributions per node, write V --------
    for (int i = tid; i < BN * BN * BN; i += THREADS) {
        const int nx = i & (BN - 1), ny = (i >> 3) & (BN - 1), nz = i >> 6;
        float acc = 0.f;
#pragma unroll
        for (int a = 0; a < 8; ++a) {
            const int dx = a & 1, dy = (a >> 1) & 1, dz = (a >> 2) & 1;
            const int le = ((nz + 1 - dz) * ET + (ny + 1 - dy)) * ET + (nx + 1 - dx);
            acc += sVeA[le * 8 + a] + sM[le] * sVeB[le * 8 + a];
        }
        V[((oz + nz) * GN + (oy + ny)) * GN + (ox + nx)] = acc;
    }
}

extern "C" void kernel_launch(void* const* d_in, const int* in_sizes, int n_in,
                              void* d_out, int out_size, void* d_ws, size_t ws_size,
                              hipStream_t stream) {
    const float* U       = (const float*)d_in[0];
    const int*   types   = (const int*)d_in[1];
    const float* filters = (const float*)d_in[2];
    float*       V       = (float*)d_out;

    dim3 grid(GN / BN, GN / BN, GN / BN);   // 16^3 = 4096 blocks
    dim3 block(THREADS);
    fe_h8_periodic_kernel<<<grid, block, 0, stream>>>(U, types, filters, V);
}